// GCN_22814866276941
// MI455X (gfx1250) — compile-verified
//
#include <hip/hip_runtime.h>
#include <math.h>

#define N_NODES 50000
#define N_EDGES 800000
#define NFEAT 512
#define NHIDDEN 128
#define NCLASS 40
#define NLAYERS 8

typedef __attribute__((ext_vector_type(2))) float v2f;
typedef __attribute__((ext_vector_type(4))) float v4f;
typedef __attribute__((ext_vector_type(8))) float v8f;
typedef __attribute__((ext_vector_type(4))) int   v4i;
typedef __attribute__((address_space(3))) v4i     lds_v4i;

// ---------------------------------------------------------------------------
// CDNA5 async global->LDS copy (ASYNCcnt path), with graceful fallback.
// Each call moves 16 bytes per lane; LDS destination address is per-lane,
// which lets us land rows with a padded stride for bank-conflict-free reads.
// Builtin expects (v4i* gaddr, AS3 v4i* ldsaddr, imm offset, imm cpol).
// ---------------------------------------------------------------------------
__device__ __forceinline__ void copy16_g2l(const float* g, float* l) {
#if __has_builtin(__builtin_amdgcn_global_load_async_to_lds_b128)
  __builtin_amdgcn_global_load_async_to_lds_b128(
      (v4i*)g, (lds_v4i*)l, 0, 0);
#else
  *(v4f*)l = *(const v4f*)g;
#endif
}

__device__ __forceinline__ void async_wait0() {
#if __has_builtin(__builtin_amdgcn_s_wait_asynccnt)
  __builtin_amdgcn_s_wait_asynccnt(0);
#else
  asm volatile("s_wait_asynccnt 0x0" ::: "memory");
#endif
}

// ---------------------------------------------------------------------------
// Dense GEMM (out = relu(A @ W (+ bias))), NOUT = 128, via V_WMMA_F32_16X16X4_F32.
// 256-thread block = 8 waves = 128 output rows. W is staged into LDS in
// 128-row chunks by async DMA and shared by all 8 waves (8x less L2 traffic).
// LDS row stride 144 floats => row k and row k+2 map to disjoint bank ranges
// for the two 16-lane halves of each ds_load_b32.
// WMMA layouts per ISA: A lanes0-15 K={k,k+1}, lanes16-31 K={k+2,k+3};
// B VGPR0=row k / k+2 per half, VGPR1=+1; D vgpr r -> rows r / r+8.
// ---------------------------------------------------------------------------
template <int K, bool BIAS, bool RELU>
__global__ __launch_bounds__(256) void gemm_wmma_kernel(
    const float* __restrict__ A, const float* __restrict__ W,
    const float* __restrict__ bias, float* __restrict__ out, int M) {
  constexpr int NOUT = 128;
  constexpr int KC = 128;        // K-chunk staged per barrier phase
  constexpr int LSTR = 144;      // padded LDS row stride (floats)
  __shared__ float ldsW[KC * LSTR];

  const int lane = threadIdx.x & 31;
  const int wave = threadIdx.x >> 5;
  const int m0 = (blockIdx.x * 8 + wave) * 16;
  const bool active = m0 < M;  // wave-uniform: EXEC stays full inside
  const int half = lane >> 4;
  const int mr = lane & 15;

  v8f acc[8] = {};
  const float* arow =
      active ? (A + (size_t)(m0 + mr) * K + 2 * half) : A;

  for (int kc = 0; kc < K; kc += KC) {
    if (kc) __syncthreads();  // previous chunk fully consumed

    // ---- stage KC x 128 chunk of W into LDS (async, padded rows) ----
    {
      const float* gw = W + (size_t)kc * NOUT;
#pragma unroll
      for (int i = 0; i < (KC * NOUT / 4) / 256; ++i) {
        const int e = threadIdx.x + 256 * i;  // float4 index
        const int r = e >> 5;                 // 32 float4 per row
        const int c4 = e & 31;
        copy16_g2l(gw + r * NOUT + c4 * 4, &ldsW[r * LSTR + c4 * 4]);
      }
    }
    async_wait0();
    __syncthreads();

    if (active) {
#pragma unroll 4
      for (int k0 = 0; k0 < KC; k0 += 4) {
        v2f a = *(const v2f*)(arow + kc + k0);  // 8B global load
        const float* wk = &ldsW[(k0 + 2 * half) * LSTR + mr];
#pragma unroll
        for (int t = 0; t < 8; ++t) {
          v2f b;
          b.x = wk[t * 16];         // LDS: W[k0+2*half  ][t*16+mr]
          b.y = wk[t * 16 + LSTR];  // LDS: W[k0+2*half+1][t*16+mr]
          acc[t] = __builtin_amdgcn_wmma_f32_16x16x4_f32(
              false, a, false, b, (short)0, acc[t], false, false);
        }
      }
    }
  }

  if (active) {
#pragma unroll
    for (int t = 0; t < 8; ++t) {
      const int n = t * 16 + mr;
      const float bv = BIAS ? bias[n] : 0.f;
      float* orow = out + (size_t)(m0 + half * 8) * NOUT + n;
#pragma unroll
      for (int r = 0; r < 8; ++r) {
        float v = acc[t][r] + bv;
        if (RELU) v = fmaxf(v, 0.f);
        orow[(size_t)r * NOUT] = v;  // row m0 + half*8 + r, col n
      }
    }
  }
}

// ---------------------------------------------------------------------------
// Zero-fill (float4 granularity) for the scatter accumulator.
// ---------------------------------------------------------------------------
__global__ __launch_bounds__(256) void zero_f4_kernel(v4f* __restrict__ p, long n4) {
  long i = (long)blockIdx.x * blockDim.x + threadIdx.x;
  if (i < n4) {
    v4f z = {0.f, 0.f, 0.f, 0.f};
    p[i] = z;
  }
}

// ---------------------------------------------------------------------------
// SpMM scatter: out[dst] += val * h[src]. One wave per edge; each lane owns
// 4 contiguous features -> coalesced b128 gather + 4 global f32 atomics.
// The 25.6MB accumulator is L2-resident (192MB L2), so atomics resolve in L2.
// ---------------------------------------------------------------------------
__global__ __launch_bounds__(256) void spmm_scatter_kernel(
    const int* __restrict__ esrc, const int* __restrict__ edst,
    const float* __restrict__ eval, const float* __restrict__ h,
    float* __restrict__ outacc) {
  const int lane = threadIdx.x & 31;
  const long e = (long)blockIdx.x * (blockDim.x / 32) + (threadIdx.x >> 5);
  if (e >= N_EDGES) return;
  const int s = esrc[e];
  const int d = edst[e];
  const float v = eval[e];
  const v4f hv = *(const v4f*)(h + (size_t)s * NHIDDEN + 4 * lane);
  float* op = outacc + (size_t)d * NHIDDEN + 4 * lane;
  atomicAdd(op + 0, v * hv.x);
  atomicAdd(op + 1, v * hv.y);
  atomicAdd(op + 2, v * hv.z);
  atomicAdd(op + 3, v * hv.w);
}

// ---------------------------------------------------------------------------
// fc1 (128 -> 40) + fused log_softmax. WMMA over 3 column tiles (48 cols).
// All out-of-range-column handling (only tile t=2) uses unconditional loads
// at clamped indices followed by selects; only the t=2 output stores are lane-masked (mr < 8).
// Row reductions on the D layout use xor-butterfly shuffles (masks 1/2/4/8
// stay inside each 16-lane half).
// ---------------------------------------------------------------------------
__global__ __launch_bounds__(128) void fc1_logsoftmax_kernel(
    const float* __restrict__ h, const float* __restrict__ W,
    const float* __restrict__ bias, float* __restrict__ out, int M) {
  const int lane = threadIdx.x & 31;
  const int wave = threadIdx.x >> 5;
  const int m0 = (blockIdx.x * 4 + wave) * 16;
  if (m0 >= M) return;
  const int half = lane >> 4;
  const int mr = lane & 15;

  v8f acc[3] = {};
  const float* arow = h + (size_t)(m0 + mr) * NHIDDEN + 2 * half;

#pragma unroll 4
  for (int k0 = 0; k0 < NHIDDEN; k0 += 4) {
    v2f a = *(const v2f*)(arow + k0);
    const int kr = k0 + 2 * half;
#pragma unroll
    for (int t = 0; t < 3; ++t) {
      const int n = t * 16 + mr;
      const bool valid = (t < 2) || (mr < 8);  // n < 40
      const int nc = valid ? n : (NCLASS - 1); // clamped, load unconditional
      const float wx = W[(size_t)kr * NCLASS + nc];
      const float wy = W[(size_t)(kr + 1) * NCLASS + nc];
      v2f b;
      b.x = valid ? wx : 0.f;
      b.y = valid ? wy : 0.f;
      acc[t] = __builtin_amdgcn_wmma_f32_16x16x4_f32(
          false, a, false, b, (short)0, acc[t], false, false);
    }
  }

  float xv[3][8];
#pragma unroll
  for (int t = 0; t < 3; ++t) {
    const bool valid = (t < 2) || (mr < 8);
    const int nb = valid ? (t * 16 + mr) : 0;  // clamped index
    const float braw = bias[nb];               // unconditional load
    const float bv = valid ? braw : 0.f;       // select, no branch
#pragma unroll
    for (int r = 0; r < 8; ++r) xv[t][r] = acc[t][r] + bv;
  }

#pragma unroll
  for (int r = 0; r < 8; ++r) {
    float mx = -INFINITY;
#pragma unroll
    for (int t = 0; t < 3; ++t)
      if ((t < 2) || (mr < 8)) mx = fmaxf(mx, xv[t][r]);
    mx = fmaxf(mx, __shfl_xor(mx, 1));
    mx = fmaxf(mx, __shfl_xor(mx, 2));
    mx = fmaxf(mx, __shfl_xor(mx, 4));
    mx = fmaxf(mx, __shfl_xor(mx, 8));

    float s = 0.f;
#pragma unroll
    for (int t = 0; t < 3; ++t)
      if ((t < 2) || (mr < 8)) s += __expf(xv[t][r] - mx);
    s += __shfl_xor(s, 1);
    s += __shfl_xor(s, 2);
    s += __shfl_xor(s, 4);
    s += __shfl_xor(s, 8);
    const float ls = __logf(s);

    const int row = m0 + half * 8 + r;
#pragma unroll
    for (int t = 0; t < 3; ++t)
      if ((t < 2) || (mr < 8))
        out[(size_t)row * NCLASS + (t * 16 + mr)] = xv[t][r] - mx - ls;
  }
}

// ---------------------------------------------------------------------------
// Host-side orchestration (all launches on `stream`, graph-capture safe).
// ---------------------------------------------------------------------------
extern "C" void kernel_launch(void* const* d_in, const int* in_sizes, int n_in,
                              void* d_out, int out_size, void* d_ws, size_t ws_size,
                              hipStream_t stream) {
  (void)in_sizes; (void)n_in; (void)out_size; (void)ws_size;
  const float* x    = (const float*)d_in[0];
  const int*   esrc = (const int*)d_in[1];
  const int*   edst = (const int*)d_in[2];
  const float* ev   = (const float*)d_in[3];
  const float* w0   = (const float*)d_in[4];
  const float* b0   = (const float*)d_in[5];
  const float* cw   = (const float*)d_in[6];
  const float* w1   = (const float*)d_in[7];
  const float* b1   = (const float*)d_in[8];
  float* out  = (float*)d_out;
  float* bufA = (float*)d_ws;                        // 50000*128 f32
  float* bufB = bufA + (size_t)N_NODES * NHIDDEN;    // 50000*128 f32

  const int nTiles = N_NODES / 16;                   // 3125 (exact)
  const int gemmBlocks = (nTiles + 7) / 8;           // 8 waves/block
  const int fc1Blocks = (nTiles + 3) / 4;            // 4 waves/block
  const long n4 = (long)N_NODES * NHIDDEN / 4;
  const int zeroBlocks = (int)((n4 + 255) / 256);
  const int spmmBlocks = (int)(((long)N_EDGES * 32 + 255) / 256);

  // h = relu(x @ fc0_w + fc0_b)
  gemm_wmma_kernel<NFEAT, true, true>
      <<<gemmBlocks, 256, 0, stream>>>(x, w0, b0, bufA, N_NODES);

  for (int i = 0; i < NLAYERS; ++i) {
    zero_f4_kernel<<<zeroBlocks, 256, 0, stream>>>((v4f*)bufB, n4);
    spmm_scatter_kernel<<<spmmBlocks, 256, 0, stream>>>(esrc, edst, ev, bufA, bufB);
    gemm_wmma_kernel<NHIDDEN, false, true>
        <<<gemmBlocks, 256, 0, stream>>>(bufB, cw + (size_t)i * NHIDDEN * NHIDDEN,
                                         nullptr, bufA, N_NODES);
  }

  // logits + log_softmax
  fc1_logsoftmax_kernel<<<fc1Blocks, 128, 0, stream>>>(bufA, w1, b1, out, N_NODES);
}